// HLD_NN1_LSTM_3264175145455
// MI455X (gfx1250) — compile-verified
//
#include <hip/hip_runtime.h>

typedef __attribute__((ext_vector_type(2))) float v2f;
typedef __attribute__((ext_vector_type(8))) float v8f;

#define B_TOT 2048
#define T_LEN 512
#define D_IN  64
#define H1    16
#define H2    4

// Fast transcendentals: v_exp_f32 + v_rcp_f32 (1 ulp rcp is fine inside sigmoid/tanh)
__device__ __forceinline__ float fast_sig(float x) {
    return __builtin_amdgcn_rcpf(1.0f + __expf(-x));
}
__device__ __forceinline__ float fast_tanh(float x) {
    // (1-e)/(1+e) = 2/(1+e) - 1,  e = exp(-2x)
    float r = __builtin_amdgcn_rcpf(1.0f + __expf(-2.0f * x));
    return __builtin_fmaf(2.0f, r, -1.0f);
}

// One wave (32 lanes) handles a 16-batch tile across all T timesteps.
// Weight-stationary: W_ih1 / W_ih2 / W_hh2 B-fragments live in VGPRs; W_hh1 in LDS.
// x A-fragments are software-pipelined one timestep ahead in registers.
__global__ __launch_bounds__(32) void HLD_NN1_LSTM_kernel(
    const float* __restrict__ x,     // [B, T, 64]
    const float* __restrict__ Wih1,  // [64, 64]
    const float* __restrict__ Whh1,  // [64, 16]
    const float* __restrict__ bih1,  // [64]
    const float* __restrict__ bhh1,  // [64]
    const float* __restrict__ Wih2,  // [16, 16]
    const float* __restrict__ Whh2,  // [16, 4]
    const float* __restrict__ bih2,  // [16]
    const float* __restrict__ bhh2,  // [16]
    const float* __restrict__ Wfc,   // [64, 4]
    const float* __restrict__ bfc,   // [64]
    float* __restrict__ out)         // [B, 64]
{
    __shared__ __align__(16) float w_hh1[64 * 16];  // W_hh1 row-major (B-frags read per step)
    __shared__ __align__(16) float h_buf[16][16];   // h1 state, batch-major
    __shared__ __align__(16) float h2_buf[16][4];   // h2 state
    __shared__ __align__(16) float g2_buf[16][16];  // lstm2 gate preacts

    const int lane = threadIdx.x;   // 0..31
    const int lm   = lane & 15;     // lane % 16
    const int lh   = lane >> 4;     // 0 or 1
    const int b0   = blockIdx.x * 16;

    // ---------------- Stage W_hh1 into LDS (coalesced float4) ----------------
    #pragma unroll
    for (int i = 0; i < 8; ++i)
        ((float4*)w_hh1)[lane + 32 * i] = ((const float4*)Whh1)[lane + 32 * i];

    // ---------------- Resident weight B-fragments ----------------
    // B-frag (16x16x4): lane holds B[K = 4q + 2*lh + {0,1}, N = lm] = W[N_row][K_col]
    v2f wih1[4][16];  // [n-tile][k-chunk]  B = W_ih1^T
    #pragma unroll
    for (int n = 0; n < 4; ++n)
        #pragma unroll
        for (int q = 0; q < 16; ++q)
            wih1[n][q] = *(const v2f*)(Wih1 + (16 * n + lm) * D_IN + 4 * q + 2 * lh);

    v2f wih2[4];      // B = W_ih2^T  (N = 16 gates, K = 16)
    #pragma unroll
    for (int q = 0; q < 4; ++q)
        wih2[q] = *(const v2f*)(Wih2 + lm * H1 + 4 * q + 2 * lh);

    v2f whh2 = *(const v2f*)(Whh2 + lm * H2 + 2 * lh);  // K = 4, one chunk

    // Per-lane gate biases: lane's D column N = lm
    float bias1[4];
    #pragma unroll
    for (int n = 0; n < 4; ++n)
        bias1[n] = bih1[16 * n + lm] + bhh1[16 * n + lm];
    const float bias2 = bih2[lm] + bhh2[lm];

    // ---------------- Zero state ----------------
    float c1[8];                    // c1[batch = r + 8*lh][hid = lm]
    #pragma unroll
    for (int r = 0; r < 8; ++r) c1[r] = 0.0f;
    float c2v[4];                   // lanes 0..15 own batch = lane
    #pragma unroll
    for (int k = 0; k < 4; ++k) c2v[k] = 0.0f;

    #pragma unroll
    for (int i = 0; i < 8; ++i) (&h_buf[0][0])[lane + 32 * i] = 0.0f;
    #pragma unroll
    for (int i = 0; i < 2; ++i) (&h2_buf[0][0])[lane + 32 * i] = 0.0f;
    __syncthreads();

    // Per-lane x row pointer: A[M = lm][K = 2*lh + ...]
    const float* px = x + (long)(b0 + lm) * T_LEN * D_IN + 2 * lh;

    // Preload A-fragments for t = 0 (pipelined one step ahead thereafter)
    v2f ax[16];
    #pragma unroll
    for (int q = 0; q < 16; ++q)
        ax[q] = *(const v2f*)(px + 4 * q);

    for (int t = 0; t < T_LEN; ++t) {
        // ---------------- LSTM1 input GEMM: x_t @ Wih1^T ----------------
        v8f acc[4];
        #pragma unroll
        for (int n = 0; n < 4; ++n)
            #pragma unroll
            for (int r = 0; r < 8; ++r) acc[n][r] = 0.0f;

        #pragma unroll
        for (int q = 0; q < 16; ++q) {
            #pragma unroll
            for (int n = 0; n < 4; ++n)
                acc[n] = __builtin_amdgcn_wmma_f32_16x16x4_f32(
                    false, ax[q], false, wih1[n][q], (short)0, acc[n], false, false);
        }

        // ax consumed: immediately issue next timestep's 16 loads (latency hidden
        // behind recurrence + gate mix + LSTM2 below).
        {
            const int tn = (t + 1 < T_LEN) ? t + 1 : t;
            const float* pxn = px + tn * D_IN;
            #pragma unroll
            for (int q = 0; q < 16; ++q)
                ax[q] = *(const v2f*)(pxn + 4 * q);
            int tp = t + 8; if (tp > T_LEN - 1) tp = T_LEN - 1;
            __builtin_prefetch(px + tp * D_IN, 0, 3);        // near-cache prefetch,
            __builtin_prefetch(px + tp * D_IN + 32, 0, 3);   // both 128B lines of row
        }

        // ---------------- recurrent GEMM: h_{t-1} @ Whh1^T (B from LDS) ----------------
        #pragma unroll
        for (int q = 0; q < 4; ++q) {
            v2f a = *(const v2f*)(&h_buf[lm][4 * q + 2 * lh]);
            #pragma unroll
            for (int n = 0; n < 4; ++n) {
                v2f b = *(const v2f*)(&w_hh1[(16 * n + lm) * H1 + 4 * q + 2 * lh]);
                acc[n] = __builtin_amdgcn_wmma_f32_16x16x4_f32(
                    false, a, false, b, (short)0, acc[n], false, false);
            }
        }

        // Gate mix (per-lane: 8 batch rows, hidden column lm)
        #pragma unroll
        for (int r = 0; r < 8; ++r) {
            float gi = fast_sig (acc[0][r] + bias1[0]);
            float gf = fast_sig (acc[1][r] + bias1[1]);
            float gg = fast_tanh(acc[2][r] + bias1[2]);
            float go = fast_sig (acc[3][r] + bias1[3]);
            float c  = __builtin_fmaf(gf, c1[r], gi * gg);
            c1[r]    = c;
            h_buf[r + 8 * lh][lm] = go * fast_tanh(c);  // raw h1_t (relu applied at read)
        }

        // ---------------- LSTM2: g2 = relu(h1_t) @ Wih2^T + h2 @ Whh2^T ----------------
        v8f acc2;
        #pragma unroll
        for (int r = 0; r < 8; ++r) acc2[r] = 0.0f;
        #pragma unroll
        for (int q = 0; q < 4; ++q) {
            v2f a;
            a.x = fmaxf(h_buf[lm][4 * q + 2 * lh],     0.0f);
            a.y = fmaxf(h_buf[lm][4 * q + 2 * lh + 1], 0.0f);
            acc2 = __builtin_amdgcn_wmma_f32_16x16x4_f32(
                false, a, false, wih2[q], (short)0, acc2, false, false);
        }
        {
            v2f a = *(const v2f*)(&h2_buf[lm][2 * lh]);
            acc2 = __builtin_amdgcn_wmma_f32_16x16x4_f32(
                false, a, false, whh2, (short)0, acc2, false, false);
        }
        #pragma unroll
        for (int r = 0; r < 8; ++r)
            g2_buf[r + 8 * lh][lm] = acc2[r] + bias2;

        // LSTM2 gate mix: lanes 0..15 each own one batch row (i=0-3,f=4-7,g=8-11,o=12-15)
        if (lane < 16) {
            #pragma unroll
            for (int k = 0; k < 4; ++k) {
                float i2 = fast_sig (g2_buf[lane][k]);
                float f2 = fast_sig (g2_buf[lane][4 + k]);
                float g2 = fast_tanh(g2_buf[lane][8 + k]);
                float o2 = fast_sig (g2_buf[lane][12 + k]);
                float c  = __builtin_fmaf(f2, c2v[k], i2 * g2);
                c2v[k]   = c;
                h2_buf[lane][k] = o2 * fast_tanh(c);
            }
        }
    }

    // ---------------- FC + sigmoid: out[b] = sig(W_fc @ h2_T + b_fc) ----------------
    float h2v[4];
    #pragma unroll
    for (int k = 0; k < 4; ++k) h2v[k] = h2_buf[lm][k];
    const int ob = 32 * lh;  // lane half covers 32 of the 64 outputs
    #pragma unroll
    for (int o = 0; o < 32; ++o) {
        const float* w = Wfc + (ob + o) * H2;
        float v = bfc[ob + o] + h2v[0] * w[0] + h2v[1] * w[1]
                              + h2v[2] * w[2] + h2v[3] * w[3];
        out[(long)(b0 + lm) * D_IN + ob + o] = fast_sig(v);
    }
}

extern "C" void kernel_launch(void* const* d_in, const int* in_sizes, int n_in,
                              void* d_out, int out_size, void* d_ws, size_t ws_size,
                              hipStream_t stream) {
    const float* x    = (const float*)d_in[0];
    const float* Wih1 = (const float*)d_in[1];
    const float* Whh1 = (const float*)d_in[2];
    const float* bih1 = (const float*)d_in[3];
    const float* bhh1 = (const float*)d_in[4];
    const float* Wih2 = (const float*)d_in[5];
    const float* Whh2 = (const float*)d_in[6];
    const float* bih2 = (const float*)d_in[7];
    const float* bhh2 = (const float*)d_in[8];
    const float* Wfc  = (const float*)d_in[9];
    const float* bfc  = (const float*)d_in[10];
    float* out = (float*)d_out;

    dim3 grid(B_TOT / 16);   // 128 single-wave workgroups, one 16-batch tile each
    dim3 block(32);
    HLD_NN1_LSTM_kernel<<<grid, block, 0, stream>>>(
        x, Wih1, Whh1, bih1, bhh1, Wih2, Whh2, bih2, bhh2, Wfc, bfc, out);
}